// PointNetFeaturePropagation_14963666059794
// MI455X (gfx1250) — compile-verified
//
#include <hip/hip_runtime.h>
#include <hip/hip_bf16.h>

// ---------------------------------------------------------------------------
// PointNet Feature Propagation on MI455X (gfx1250, wave32)
//   Stage 1: 3-NN inverse-distance interpolation (async-LDS-staged xyz2 scan)
//   Stage 2: GEMM1 (384->256) via V_WMMA_F32_16X16X4_F32, LDS-tiled B operand
//   Stage 3: BN stats (per-channel mean/var over B*N) -> scale/shift
//   Stage 4: BN+ReLU elementwise (in place)
//   Stage 5: GEMM2 (256->128) via WMMA -> d_out (pre-BN)
//   Stage 6: BN stats + BN+ReLU in place on d_out
// ---------------------------------------------------------------------------

#define Bb   4
#define Nn   16384
#define Ss   4096
#define D1c  128
#define D2c  256
#define M1   256
#define K1   384
#define M2   128
#define K2   256

typedef float v2f __attribute__((ext_vector_type(2)));
typedef float v4f __attribute__((ext_vector_type(4)));
typedef float v8f __attribute__((ext_vector_type(8)));
typedef int   v4i __attribute__((ext_vector_type(4)));

// ---- gfx1250 async global->LDS staging (ASYNCcnt path), with safe fallback --
#if defined(__has_builtin)
#if __has_builtin(__builtin_amdgcn_global_load_async_to_lds_b128) && \
    __has_builtin(__builtin_amdgcn_s_wait_asynccnt)
#define USE_ASYNC_LDS 1
#endif
#endif
#ifndef USE_ASYNC_LDS
#define USE_ASYNC_LDS 0
#endif

__device__ __forceinline__ void stage16(const float* __restrict__ src, float* dst) {
#if USE_ASYNC_LDS
  // builtin signature: (v4i addrspace(1)* src, v4i addrspace(3)* dst, Ii, Ii)
  __builtin_amdgcn_global_load_async_to_lds_b128(
      (__attribute__((address_space(1))) v4i*)(void*)src,
      (__attribute__((address_space(3))) v4i*)(void*)dst, 0, 0);
#else
  *(v4f*)dst = *(const v4f*)src;
#endif
}

__device__ __forceinline__ void stage_fence() {
#if USE_ASYNC_LDS
  __builtin_amdgcn_s_wait_asynccnt(0);   // s_wait_asynccnt 0
#endif
  __syncthreads();
}

// ---------------------------------------------------------------------------
// Stage 1: kNN(3) + inverse-distance-weighted interpolation.
// One thread per query point; xyz2[b] (3 x 4096 = 48KB) staged in LDS so the
// inner scan is LDS-broadcast reads (all lanes read the same s).
// ---------------------------------------------------------------------------
__global__ void knn_interp_kernel(const float* __restrict__ xyz1,
                                  const float* __restrict__ xyz2,
                                  const float* __restrict__ p2,
                                  float* __restrict__ interp) {
  __shared__ float sx[3 * Ss];   // 48 KB of the 320 KB/WGP LDS
  const int b   = blockIdx.y;
  const int tid = threadIdx.x;

  const float* x2 = xyz2 + (size_t)b * 3 * Ss;
#pragma unroll
  for (int q = 0; q < 12; ++q) {          // 3*4096 floats = 256 thr * 12 * v4f
    const int i = (q * 256 + tid) * 4;
    stage16(x2 + i, &sx[i]);
  }
  stage_fence();

  const int n = blockIdx.x * blockDim.x + tid;
  const float px = xyz1[((size_t)b * 3 + 0) * Nn + n];
  const float py = xyz1[((size_t)b * 3 + 1) * Nn + n];
  const float pz = xyz1[((size_t)b * 3 + 2) * Nn + n];

  float d0 = 3.0e37f, d1 = 3.0e37f, d2 = 3.0e37f;
  int   i0 = 0,       i1 = 0,       i2 = 0;

  for (int s = 0; s < Ss; ++s) {
    const float dx = px - sx[s];
    const float dy = py - sx[Ss + s];
    const float dz = pz - sx[2 * Ss + s];
    const float d  = fmaf(dx, dx, fmaf(dy, dy, dz * dz));
    if (d < d2) {
      if (d < d1) {
        d2 = d1; i2 = i1;
        if (d < d0) { d1 = d0; i1 = i0; d0 = d; i0 = s; }
        else        { d1 = d;  i1 = s; }
      } else        { d2 = d;  i2 = s; }
    }
  }

  const float r0 = 1.0f / (d0 + 1e-8f);
  const float r1 = 1.0f / (d1 + 1e-8f);
  const float r2 = 1.0f / (d2 + 1e-8f);
  const float inv = 1.0f / (r0 + r1 + r2);
  const float w0 = (d0 > 1e8f) ? 0.0f : r0 * inv;
  const float w1 = (d1 > 1e8f) ? 0.0f : r1 * inv;
  const float w2 = (d2 > 1e8f) ? 0.0f : r2 * inv;

  const float* p2b  = p2 + (size_t)b * D2c * Ss;
  float*       outb = interp + (size_t)b * D2c * Nn + n;
  for (int c = 0; c < D2c; ++c) {
    const float* row = p2b + (size_t)c * Ss;     // gathers stay L2-resident
    const float v = fmaf(w0, row[i0], fmaf(w1, row[i1], w2 * row[i2]));
    outb[(size_t)c * Nn] = v;
  }
}

// ---------------------------------------------------------------------------
// Stage 2/5: fp32 GEMM via V_WMMA_F32_16X16X4_F32.
// Y[b][m][n] = sum_k W[m][k] * X[b][k][n] + bias[m]
// X row k comes from X0 (k < C0) or X1 (k >= C0): the concat select happens
// only in the (uniform-flow) LDS staging phase, never in the WMMA loop.
// Block = 8 waves = 128(M) x 64(N) tile; each wave owns one 16x64 strip
// (4 accumulators sharing one A fragment per K-step). B operand staged in
// LDS per 32-deep K-chunk -> 8x less global traffic, no exec-mask juggling.
// ---------------------------------------------------------------------------
#define KC   32
#define LSTR 72   // 64 cols + 8 pad: K-half row groups hit disjoint LDS banks

__global__ void gemm_wmma_f32(const float* __restrict__ W,
                              const float* __restrict__ bias,
                              const float* __restrict__ X0,
                              const float* __restrict__ X1,
                              float* __restrict__ Y,
                              int M, int K, int C0) {
  __shared__ float sx[KC * LSTR];   // 9 KB
  const int tid  = threadIdx.x;
  const int lane = tid & 31;
  const int wave = tid >> 5;
  const int b    = blockIdx.z;
  const int m0   = (blockIdx.y * 8 + wave) * 16;
  const int n0   = blockIdx.x * 64;
  const int half = lane >> 4;      // K-half this lane carries (ISA A/B layout)
  const int l15  = lane & 15;

  const float* wr  = W + (size_t)(m0 + l15) * K;          // A rows, 8B aligned
  const float* Xb0 = X0 + (size_t)b * C0 * Nn + n0;
  const float* Xb1 = X1 + (size_t)b * (K - C0) * Nn + n0;

  v8f acc0 = {}, acc1 = {}, acc2 = {}, acc3 = {};

  for (int k0 = 0; k0 < K; k0 += KC) {
    // ---- stage X[k0..k0+KC-1][n0..n0+63] into LDS (512 v4f, 2 per thread) --
#pragma unroll
    for (int q = 0; q < 2; ++q) {
      const int idx = tid * 2 + q;            // 0..511
      const int r   = idx >> 4;               // k-row within chunk
      const int c4  = (idx & 15) << 2;        // col (float4 granular)
      const int k   = k0 + r;
      const float* src = (k < C0) ? (Xb0 + (size_t)k * Nn + c4)
                                  : (Xb1 + (size_t)(k - C0) * Nn + c4);
      stage16(src, &sx[r * LSTR + c4]);
    }
    stage_fence();

    // ---- 8 K-steps of 4 WMMAs each, B from LDS, A from global (L1-hot) ----
#pragma unroll
    for (int kk = 0; kk < KC; kk += 4) {
      const v2f a = *(const v2f*)(wr + k0 + kk + 2 * half);  // global_load_b64
      const float* s0 = &sx[(kk + 2 * half) * LSTR + l15];
      const float* s1 = s0 + LSTR;
      v2f b0; b0.x = s0[0];  b0.y = s1[0];
      v2f b1; b1.x = s0[16]; b1.y = s1[16];
      v2f b2; b2.x = s0[32]; b2.y = s1[32];
      v2f b3; b3.x = s0[48]; b3.y = s1[48];
      acc0 = __builtin_amdgcn_wmma_f32_16x16x4_f32(false, a, false, b0, (short)0, acc0, false, false);
      acc1 = __builtin_amdgcn_wmma_f32_16x16x4_f32(false, a, false, b1, (short)0, acc1, false, false);
      acc2 = __builtin_amdgcn_wmma_f32_16x16x4_f32(false, a, false, b2, (short)0, acc2, false, false);
      acc3 = __builtin_amdgcn_wmma_f32_16x16x4_f32(false, a, false, b3, (short)0, acc3, false, false);
    }
    __syncthreads();   // protect LDS tile before next chunk's staging (WAR)
  }

  // C/D layout: element (vgpr r, lane L): m = m0 + r + 8*(L>>4), n = n0+16j+(L&15)
  v8f accs[4] = {acc0, acc1, acc2, acc3};
#pragma unroll
  for (int j = 0; j < 4; ++j) {
    const int n = n0 + 16 * j + l15;
#pragma unroll
    for (int r = 0; r < 8; ++r) {
      const int m = m0 + r + 8 * half;
      Y[((size_t)b * M + m) * Nn + n] = accs[j][r] + bias[m];
    }
  }
}

// ---------------------------------------------------------------------------
// Stage 3/6a: per-channel BN statistics over (B, N); fold gamma/beta into
// scale/shift so the elementwise pass is a single FMA + ReLU.
// ---------------------------------------------------------------------------
__global__ void bn_stats_kernel(const float* __restrict__ H,
                                const float* __restrict__ g,
                                const float* __restrict__ be,
                                float* __restrict__ scale,
                                float* __restrict__ shift,
                                int C) {
  __shared__ float ssum[256];
  __shared__ float ssq[256];
  const int c   = blockIdx.x;
  const int tid = threadIdx.x;
  const unsigned total = Bb * Nn;

  float sum = 0.0f, sq = 0.0f;
  for (unsigned i = tid; i < total; i += blockDim.x) {
    const unsigned bb = i / Nn;
    const unsigned nn = i % Nn;        // Nn = 2^14 -> strength-reduced
    const float v = H[((size_t)bb * C + c) * Nn + nn];
    sum += v;
    sq = fmaf(v, v, sq);
  }
  ssum[tid] = sum; ssq[tid] = sq;
  __syncthreads();
  for (int off = 128; off > 0; off >>= 1) {
    if (tid < off) { ssum[tid] += ssum[tid + off]; ssq[tid] += ssq[tid + off]; }
    __syncthreads();
  }
  if (tid == 0) {
    const float inv_n = 1.0f / (float)total;
    const float mean  = ssum[0] * inv_n;
    const float var   = ssq[0] * inv_n - mean * mean;   // biased, torch BN
    const float a     = g[c] * rsqrtf(var + 1e-5f);
    scale[c] = a;
    shift[c] = be[c] - mean * a;
  }
}

__global__ void bn_relu_kernel(float* __restrict__ H,
                               const float* __restrict__ scale,
                               const float* __restrict__ shift,
                               int C, unsigned total) {
  const unsigned i = blockIdx.x * blockDim.x + threadIdx.x;
  if (i >= total) return;
  const int c = (int)((i / (unsigned)Nn) % (unsigned)C);
  H[i] = fmaxf(fmaf(H[i], scale[c], shift[c]), 0.0f);
}

// ---------------------------------------------------------------------------
extern "C" void kernel_launch(void* const* d_in, const int* in_sizes, int n_in,
                              void* d_out, int out_size, void* d_ws, size_t ws_size,
                              hipStream_t stream) {
  const float* xyz1    = (const float*)d_in[0];
  const float* xyz2    = (const float*)d_in[1];
  const float* points1 = (const float*)d_in[2];
  const float* points2 = (const float*)d_in[3];
  // d_in[4], d_in[5]: idx1/idx2 (pure batch ids -> implicit in our layout)
  const float* w1  = (const float*)d_in[6];
  const float* b1  = (const float*)d_in[7];
  const float* g1  = (const float*)d_in[8];
  const float* be1 = (const float*)d_in[9];
  const float* w2  = (const float*)d_in[10];
  const float* b2  = (const float*)d_in[11];
  const float* g2  = (const float*)d_in[12];
  const float* be2 = (const float*)d_in[13];
  float* out = (float*)d_out;

  float* ws     = (float*)d_ws;
  float* interp = ws;                                   // B*D2*N
  float* h1     = interp + (size_t)Bb * D2c * Nn;       // B*M1*N
  float* scale1 = h1 + (size_t)Bb * M1 * Nn;            // M1
  float* shift1 = scale1 + M1;
  float* scale2 = shift1 + M1;                          // M2
  float* shift2 = scale2 + M2;

  // Stage 1: 3-NN interpolation
  knn_interp_kernel<<<dim3(Nn / 256, Bb), 256, 0, stream>>>(xyz1, xyz2, points2, interp);

  // Stage 2: GEMM1 (concat(points1, interp) -> 256 channels) + bias
  gemm_wmma_f32<<<dim3(Nn / 64, M1 / 128, Bb), 256, 0, stream>>>(
      w1, b1, points1, interp, h1, M1, K1, D1c);

  // Stage 3/4: BN1 stats + BN-ReLU in place on h1
  bn_stats_kernel<<<M1, 256, 0, stream>>>(h1, g1, be1, scale1, shift1, M1);
  {
    const unsigned total = (unsigned)Bb * M1 * Nn;
    bn_relu_kernel<<<(total + 255) / 256, 256, 0, stream>>>(h1, scale1, shift1, M1, total);
  }

  // Stage 5: GEMM2 -> d_out (pre-BN)
  gemm_wmma_f32<<<dim3(Nn / 64, M2 / 128, Bb), 256, 0, stream>>>(
      w2, b2, h1, h1, out, M2, K2, K2);

  // Stage 6: BN2 stats + BN-ReLU in place on d_out
  bn_stats_kernel<<<M2, 256, 0, stream>>>(out, g2, be2, scale2, shift2, M2);
  {
    const unsigned total = (unsigned)Bb * M2 * Nn;
    bn_relu_kernel<<<(total + 255) / 256, 256, 0, stream>>>(out, scale2, shift2, M2, total);
  }
}